// PhysNetInteractionLayer_4810363372620
// MI455X (gfx1250) — compile-verified
//
#include <hip/hip_runtime.h>
#include <hip/hip_bf16.h>

#define FDIM   128
#define NNODE  32768
#define MNBR   32
#define KDIM   64
#define NPB    4            // nodes per block in node_attention
#define LOG2F_ 0.6931471805599453f

typedef float v2f_t __attribute__((ext_vector_type(2)));
typedef float v8f_t __attribute__((ext_vector_type(8)));

__device__ __forceinline__ float ssp(float a) {
    // shifted softplus: softplus(a) - ln2, numerically stable
    return fmaxf(a, 0.0f) + log1pf(__expf(-fabsf(a))) - LOG2F_;
}

__device__ __forceinline__ v8f_t wmma4(v2f_t a, v2f_t b, v8f_t c) {
    // V_WMMA_F32_16X16X4_F32 : D = A(16x4) * B(4x16) + C(16x16), exact fp32
    return __builtin_amdgcn_wmma_f32_16x16x4_f32(
        false, a, false, b, (short)0, c, false, false);
}

// ---------------------------------------------------------------------------
// Generic fused GEMM:  out[N,128] = epilogue( act(A)[N,128] @ W[128,128]^T )
// MODE 0: out = acc + bias
// MODE 1: out = acc + bias + add[row,col]          (residual, safe in-place)
// MODE 2: out = acc + bias + u[col]*xin[row,col]   (final layer)
// SSP_A : apply shifted-softplus to A elements on load
// Block: 256 thr = 8 waves; tile 64 rows x 128 cols; wave = 16 rows x 64 cols
// ---------------------------------------------------------------------------
template <bool SSP_A, int MODE>
__global__ __launch_bounds__(256) void gemm_fused(
    const float* __restrict__ A, const float* __restrict__ W,
    const float* __restrict__ bias, const float* __restrict__ add,
    const float* __restrict__ xin, const float* __restrict__ u,
    float* __restrict__ out)
{
    const int lane = threadIdx.x & 31;
    const int wave = threadIdx.x >> 5;
    const int rw = wave & 3;        // row-tile group within block (4)
    const int cw = wave >> 2;       // col half (2)
    const int rowBase = blockIdx.x * 64 + rw * 16;
    const int colBase = cw * 64;
    const int lr = lane & 15;       // M (A) / N (B,C) index within tile
    const int hk = lane >> 4;       // K sub-offset selector (0/1 -> +0/+2)

    const float* Arow = A + (size_t)(rowBase + lr) * FDIM;

    v8f_t acc[4] = {v8f_t{}, v8f_t{}, v8f_t{}, v8f_t{}};

    for (int k = 0; k < FDIM; k += 4) {
        const int kk = k + 2 * hk;
        v2f_t a;
        a.x = Arow[kk];
        a.y = Arow[kk + 1];
        if (SSP_A) { a.x = ssp(a.x); a.y = ssp(a.y); }
#pragma unroll
        for (int t = 0; t < 4; ++t) {
            const float* Wr = W + (size_t)(colBase + t * 16 + lr) * FDIM;
            v2f_t b;
            b.x = Wr[kk];
            b.y = Wr[kk + 1];
            acc[t] = wmma4(a, b, acc[t]);
        }
    }

#pragma unroll
    for (int t = 0; t < 4; ++t) {
        const int col = colBase + t * 16 + lr;
        const float bv = bias[col];
#pragma unroll
        for (int v = 0; v < 8; ++v) {
            const int row = rowBase + v + 8 * hk;   // C layout: hi lanes -> M+8
            float r = acc[t][v] + bv;
            if (MODE == 1) r += add[(size_t)row * FDIM + col];
            if (MODE == 2) r += u[col] * xin[(size_t)row * FDIM + col];
            out[(size_t)row * FDIM + col] = r;
        }
    }
}

// ---------------------------------------------------------------------------
// Fused per-node kernel: filter GEMM + gather + faithful-reshape attention.
// One block (128 thr / 4 waves) processes NPB nodes; the k2f_w B-fragments
// for this wave's 32-column slice are hoisted into registers ONCE per block
// (32 x v2f = 64 VGPRs), so the per-node K-loop is LDS-A x register-B only.
// Per node n:
//   g[32,128]  = rbf[n*32 .. n*32+31, 0:64] @ k2f_w^T        (WMMA f32)
//   xj[m,f]    = g[m,f] * xjw[idx_j[n*32+m], f]              (L2 gather)
//   l[m]       = sum_f xi[n,f] * xj_flat[f*32+m]             (reshape trick)
//   att        = softmax_m(l);  m0[f] = xi[n,f] + sum_m att[m]*xj[m,f]
// rbf is the one compulsory HBM stream: staged with non-temporal loads
// (keep L2 for xjw gathers + weights) and next-tile global_prefetch.
// ---------------------------------------------------------------------------
#define RBF_STRIDE 68   // 64 + 4 pad -> conflict-free A-fragment reads
#define XJ_STRIDE 130   // 128 + 2 pad

__global__ __launch_bounds__(128) void node_attention(
    const float* __restrict__ rbf,    // [E,64]
    const int*   __restrict__ idx_j,  // [E]
    const float* __restrict__ k2f_w,  // [128,64]
    const float* __restrict__ xi,     // [N,128]
    const float* __restrict__ xjw,    // [N,128]
    float* __restrict__ mout)         // [N,128]
{
    __shared__ float s_rbf[MNBR * RBF_STRIDE];
    __shared__ float s_xj[MNBR * XJ_STRIDE];
    __shared__ float s_xi[FDIM];
    __shared__ float s_att[MNBR];
    __shared__ int   s_idx[MNBR];

    const int tid  = threadIdx.x;
    const int lane = tid & 31;
    const int wave = tid >> 5;
    const int lr   = lane & 15;
    const int hk   = lane >> 4;

    // Hoist this wave's k2f_w fragments into registers (reused for NPB nodes).
    v2f_t bfrag[2][KDIM / 4];
#pragma unroll
    for (int t = 0; t < 2; ++t) {
        const float* Wr = k2f_w + (size_t)(wave * 32 + t * 16 + lr) * KDIM;
#pragma unroll
        for (int ks = 0; ks < KDIM / 4; ++ks) {
            const int kk = ks * 4 + 2 * hk;
            bfrag[t][ks].x = Wr[kk];
            bfrag[t][ks].y = Wr[kk + 1];
        }
    }

    for (int i = 0; i < NPB; ++i) {
        const int n = blockIdx.x * NPB + i;
        const float* rbf_base = rbf + (size_t)n * MNBR * KDIM;

        // Prefetch next node's 8KB rbf tile (64B/thread covers it).
        if (i + 1 < NPB)
            __builtin_prefetch(rbf_base + MNBR * KDIM + tid * 16, 0, 1);

        // Stage rbf tile (non-temporal: one-shot stream), idx row, xi row.
        for (int q = tid; q < MNBR * KDIM; q += 128) {
            s_rbf[(q >> 6) * RBF_STRIDE + (q & 63)] =
                __builtin_nontemporal_load(rbf_base + q);
        }
        if (tid < MNBR) s_idx[tid] = idx_j[(size_t)n * MNBR + tid];
        s_xi[tid] = xi[(size_t)n * FDIM + tid];
        __syncthreads();

        // GEMM: [32,64] x [64,128]; wave handles cols [32*wave, 32*wave+32).
        v8f_t acc[2][2] = {{v8f_t{}, v8f_t{}}, {v8f_t{}, v8f_t{}}};
#pragma unroll
        for (int ks = 0; ks < KDIM / 4; ++ks) {
            const int kk = ks * 4 + 2 * hk;
            v2f_t a0, a1;
            a0.x = s_rbf[lr * RBF_STRIDE + kk];
            a0.y = s_rbf[lr * RBF_STRIDE + kk + 1];
            a1.x = s_rbf[(16 + lr) * RBF_STRIDE + kk];
            a1.y = s_rbf[(16 + lr) * RBF_STRIDE + kk + 1];
#pragma unroll
            for (int t = 0; t < 2; ++t) {
                acc[0][t] = wmma4(a0, bfrag[t][ks], acc[0][t]);
                acc[1][t] = wmma4(a1, bfrag[t][ks], acc[1][t]);
            }
        }

        // Filter by gathered xjw rows (L2-resident), deposit xj tile to LDS.
#pragma unroll
        for (int r = 0; r < 2; ++r) {
#pragma unroll
            for (int t = 0; t < 2; ++t) {
#pragma unroll
                for (int v = 0; v < 8; ++v) {
                    const int m = r * 16 + v + 8 * hk;
                    const int f = wave * 32 + t * 16 + lr;
                    const int j = s_idx[m];
                    s_xj[m * XJ_STRIDE + f] =
                        acc[r][t][v] * xjw[(size_t)j * FDIM + f];
                }
            }
        }
        __syncthreads();

        // Attention logits over the faithful [N,F,M] reshape; softmax, wave 0.
        if (wave == 0) {
            float l = 0.0f;
#pragma unroll 8
            for (int f = 0; f < FDIM; ++f) {
                const int flat = f * MNBR + lane;  // index into flat 32x128 blk
                l += s_xi[f] * s_xj[(flat >> 7) * XJ_STRIDE + (flat & 127)];
            }
            float mx = l;
#pragma unroll
            for (int o = 16; o > 0; o >>= 1)
                mx = fmaxf(mx, __shfl_xor(mx, o, 32));
            const float e = __expf(l - mx);
            float sum = e;
#pragma unroll
            for (int o = 16; o > 0; o >>= 1)
                sum += __shfl_xor(sum, o, 32);
            s_att[lane] = e / sum;
        }
        __syncthreads();

        // m0[f] = xi[f] + sum_m att[m] * xj[m,f]   (thread per f)
        float accv = s_xi[tid];
#pragma unroll 8
        for (int m = 0; m < MNBR; ++m)
            accv += s_att[m] * s_xj[m * XJ_STRIDE + tid];
        mout[(size_t)n * FDIM + tid] = accv;

        __syncthreads();   // protect LDS reuse across the node loop
    }
}

// ---------------------------------------------------------------------------
extern "C" void kernel_launch(void* const* d_in, const int* in_sizes, int n_in,
                              void* d_out, int out_size, void* d_ws, size_t ws_size,
                              hipStream_t stream)
{
    const float* x      = (const float*)d_in[0];
    const float* rbf    = (const float*)d_in[1];
    // d_in[2] = idx_i : unused by the reference computation
    const int*   idx_j  = (const int*)d_in[3];
    const float* k2f_w  = (const float*)d_in[4];
    const float* wi     = (const float*)d_in[5];
    const float* bi     = (const float*)d_in[6];
    const float* wj     = (const float*)d_in[7];
    const float* bj     = (const float*)d_in[8];
    const float* res_w1 = (const float*)d_in[9];
    const float* res_b1 = (const float*)d_in[10];
    const float* res_w2 = (const float*)d_in[11];
    const float* res_b2 = (const float*)d_in[12];
    const float* wd     = (const float*)d_in[13];
    const float* bd     = (const float*)d_in[14];
    const float* u      = (const float*)d_in[15];
    float* out = (float*)d_out;

    const size_t NF = (size_t)NNODE * FDIM;
    float* xi_buf  = (float*)d_ws;
    float* xjw_buf = xi_buf + NF;
    float* m_buf   = xjw_buf + NF;
    float* t_buf   = m_buf + NF;

    dim3 gblk(256), ggrd(NNODE / 64);

    // xi = ssp(x)@wi^T + bi ; xjw = ssp(x)@wj^T + bj
    gemm_fused<true, 0><<<ggrd, gblk, 0, stream>>>(x, wi, bi, nullptr, nullptr, nullptr, xi_buf);
    gemm_fused<true, 0><<<ggrd, gblk, 0, stream>>>(x, wj, bj, nullptr, nullptr, nullptr, xjw_buf);

    // fused filter-GEMM + gather + attention -> m
    node_attention<<<NNODE / NPB, 128, 0, stream>>>(rbf, idx_j, k2f_w, xi_buf, xjw_buf, m_buf);

    // two residual blocks: m += (ssp(m)@w1^T + b1)@w2^T + b2
    for (int r = 0; r < 2; ++r) {
        const float* w1 = res_w1 + (size_t)r * FDIM * FDIM;
        const float* b1 = res_b1 + (size_t)r * FDIM;
        const float* w2 = res_w2 + (size_t)r * FDIM * FDIM;
        const float* b2 = res_b2 + (size_t)r * FDIM;
        gemm_fused<true, 0><<<ggrd, gblk, 0, stream>>>(m_buf, w1, b1, nullptr, nullptr, nullptr, t_buf);
        gemm_fused<false, 1><<<ggrd, gblk, 0, stream>>>(t_buf, w2, b2, m_buf, nullptr, nullptr, m_buf);
    }

    // out = u*x + ssp(m)@wd^T + bd
    gemm_fused<true, 2><<<ggrd, gblk, 0, stream>>>(m_buf, wd, bd, nullptr, x, u, out);
}